// DiffusionAnomalyAttention_22660247454299
// MI455X (gfx1250) — compile-verified
//
#include <hip/hip_runtime.h>
#include <math.h>

// B=4, L=1024, H=8, E=64, D=64
#define Bc 4
#define Lc 1024
#define Hc 8
#define Ec 64
#define Dc 64

typedef __attribute__((ext_vector_type(2))) float v2f;
typedef __attribute__((ext_vector_type(8))) float v8f;

__device__ __forceinline__ v8f wmma4(v2f a, v2f b, v8f c) {
  // V_WMMA_F32_16X16X4_F32 : D = A(16x4) * B(4x16) + C(16x16), all f32
  return __builtin_amdgcn_wmma_f32_16x16x4_f32(false, a, false, b, (short)0, c,
                                               false, false);
}

// LDS layout (floats)
#define PPITCH 1028                 // 16 rows x 1028 (pad 4 to dodge bank conflicts)
#define PO (16 * PPITCH)            // phase-2 partial-output area: 2*4*256 = 2048
#define RO (PO + 2048)              // per-wave row-max: 8 waves x 16 rows
#define SOFS (RO + 128)             // per-wave row-sum: 8 waves x 16 rows
#define SMEM_FLOATS (SOFS + 128)    // 18752 floats = 75008 bytes (< 320KB/WGP)

__global__ __launch_bounds__(256) void attn_kernel(
    const float* __restrict__ Q, const float* __restrict__ K,
    const float* __restrict__ Vv, float* __restrict__ outV,
    float* __restrict__ outS) {
  __shared__ __align__(16) float smem[SMEM_FLOATS];

  const int tid  = threadIdx.x;
  const int w    = tid >> 5;     // wave id 0..7 (wave32)
  const int lane = tid & 31;
  const int lr   = lane & 15;    // row (A) / col (B,C,D) inside a 16-wide tile
  const int hi   = lane >> 4;    // half-wave select
  const int eoff = hi * 2;       // K-offset for A/B fragments of 16x16x4

  const int blk = blockIdx.x;    // ((b*H + h)*64 + mtile)
  const int mt  = blk & 63;
  const int bh  = blk >> 6;
  const int h   = bh & 7;
  const int b   = bh >> 3;
  const int m0  = mt * 16;

  // ---- preload Q A-fragments (16 k-steps covering E=64) -------------------
  const float* qrow = Q + (((size_t)b * Lc + (m0 + lr)) * Hc + h) * Ec + eoff;
  v2f aq[16];
#pragma unroll
  for (int kk = 0; kk < 16; ++kk) aq[kk] = *(const v2f*)(qrow + 4 * kk);

  // ---- scores: each wave owns 128 columns = 8 tiles -----------------------
  v8f acc[8];
#pragma unroll
  for (int t = 0; t < 8; ++t) acc[t] = v8f{0, 0, 0, 0, 0, 0, 0, 0};

#pragma unroll
  for (int t = 0; t < 8; ++t) {
    const int n0 = w * 128 + t * 16;
    const float* kr =
        K + (((size_t)b * Lc + (n0 + lr)) * Hc + h) * Ec + eoff;
#pragma unroll
    for (int kk = 0; kk < 16; ++kk) {
      v2f bf = *(const v2f*)(kr + 4 * kk);
      acc[t] = wmma4(aq[kk], bf, acc[t]);
    }
  }

  const float scale = 0.125f;  // 1/sqrt(64)

  // ---- row max (masked): lanes -> half-wave shuffle -> LDS across waves ---
  float pm[8];
#pragma unroll
  for (int r = 0; r < 8; ++r) pm[r] = -INFINITY;
#pragma unroll
  for (int t = 0; t < 8; ++t) {
    const int n = w * 128 + t * 16 + lr;
#pragma unroll
    for (int r = 0; r < 8; ++r) {
      const int m = m0 + r + 8 * hi;
      if (n <= m) pm[r] = fmaxf(pm[r], scale * acc[t][r]);
    }
  }
#pragma unroll
  for (int d = 1; d < 16; d <<= 1)
#pragma unroll
    for (int r = 0; r < 8; ++r)
      pm[r] = fmaxf(pm[r], __shfl_xor(pm[r], d, 32));
  if (lr == 0) {
#pragma unroll
    for (int r = 0; r < 8; ++r) smem[RO + w * 16 + 8 * hi + r] = pm[r];
  }
  __syncthreads();
  float rm[8];
#pragma unroll
  for (int r = 0; r < 8; ++r) {
    float mx = smem[RO + 8 * hi + r];
#pragma unroll
    for (int ww = 1; ww < 8; ++ww)
      mx = fmaxf(mx, smem[RO + ww * 16 + 8 * hi + r]);
    rm[r] = mx;
  }

  // ---- exp + row sum ------------------------------------------------------
  float ps[8];
#pragma unroll
  for (int r = 0; r < 8; ++r) ps[r] = 0.0f;
#pragma unroll
  for (int t = 0; t < 8; ++t) {
    const int n = w * 128 + t * 16 + lr;
#pragma unroll
    for (int r = 0; r < 8; ++r) {
      const int m = m0 + r + 8 * hi;
      float p = (n <= m) ? __expf(scale * acc[t][r] - rm[r]) : 0.0f;
      acc[t][r] = p;
      ps[r] += p;
    }
  }
#pragma unroll
  for (int d = 1; d < 16; d <<= 1)
#pragma unroll
    for (int r = 0; r < 8; ++r) ps[r] += __shfl_xor(ps[r], d, 32);
  if (lr == 0) {
#pragma unroll
    for (int r = 0; r < 8; ++r) smem[SOFS + w * 16 + 8 * hi + r] = ps[r];
  }
  __syncthreads();
  float inv[8];
#pragma unroll
  for (int r = 0; r < 8; ++r) {
    float s = 0.0f;
#pragma unroll
    for (int ww = 0; ww < 8; ++ww) s += smem[SOFS + ww * 16 + 8 * hi + r];
    inv[r] = 1.0f / s;
  }

  // ---- normalize, write series, stage P into LDS --------------------------
  float* outSrow = outS + (size_t)bh * Lc * Lc;
#pragma unroll
  for (int t = 0; t < 8; ++t) {
    const int n = w * 128 + t * 16 + lr;
#pragma unroll
    for (int r = 0; r < 8; ++r) {
      const int mloc = r + 8 * hi;
      const float val = acc[t][r] * inv[r];
      outSrow[(size_t)(m0 + mloc) * Lc + n] = val;
      smem[mloc * PPITCH + n] = val;
    }
  }
  __syncthreads();

  // ---- phase 2: V_tile(16x64) = P(16x1024) * values(1024x64) --------------
  // wave w: output tile tt = w&3 (d range tt*16..), K-half kh = w>>2
  const int tt = w & 3;
  const int kh = w >> 2;
  const int kstart = kh * 512;
  const float* vbase =
      Vv + (size_t)b * Lc * Hc * Dc + (size_t)h * Dc + tt * 16 + lr;
  const float* prow = &smem[lr * PPITCH];

  v8f vacc = v8f{0, 0, 0, 0, 0, 0, 0, 0};
  for (int ks = 0; ks < 128; ++ks) {
    const int k = kstart + ks * 4 + eoff;
    v2f af = *(const v2f*)(prow + k);            // A = P rows, ds_load_b64
    v2f bf;
    bf.x = vbase[(size_t)k * (Hc * Dc)];         // B = values rows k, k+1
    bf.y = vbase[(size_t)(k + 1) * (Hc * Dc)];
    vacc = wmma4(af, bf, vacc);
  }

  float* pout = &smem[PO + (kh * 4 + tt) * 256];
#pragma unroll
  for (int r = 0; r < 8; ++r) pout[(r + 8 * hi) * 16 + lr] = vacc[r];
  __syncthreads();

  // ---- reduce the two K-halves and store V [B,L,H,D] ----------------------
#pragma unroll
  for (int i = 0; i < 4; ++i) {
    const int idx = tid + i * 256;        // 16 rows x 64 cols
    const int m  = idx >> 6;
    const int d  = idx & 63;
    const int t2 = d >> 4;
    const int n  = d & 15;
    const float val = smem[PO + t2 * 256 + m * 16 + n] +
                      smem[PO + (4 + t2) * 256 + m * 16 + n];
    outV[(((size_t)b * Lc + (m0 + m)) * Hc + h) * Dc + d] = val;
  }
}

// ---- prior / sigma_t: pure streaming stores -------------------------------
__global__ __launch_bounds__(256) void prior_kernel(
    const float* __restrict__ sigma, const float* __restrict__ sac,
    const int* __restrict__ tstep, float* __restrict__ outP,
    float* __restrict__ outT) {
  const int row = blockIdx.x;        // (b*H + h)*L + l
  const int l  = row & (Lc - 1);
  const int bh = row >> 10;
  const int h  = bh & 7;
  const int b  = bh >> 3;

  const float x  = sigma[((size_t)b * Lc + l) * Hc + h];
  const float sg = 1.0f / (1.0f + __expf(-5.0f * x)) + 1e-5f;
  const float sval =
      (__expf(sg * 1.0986122886681098f) - 1.0f) * sac[tstep[b]];  // 3^sg - 1
  const float amp = 0.3989422804014327f / sval;   // 1/sqrt(2*pi) / sigma_t
  const float cf  = -0.5f / (sval * sval);

  const size_t base = ((size_t)bh * Lc + l) * Lc;
  const int s0 = threadIdx.x * 4;

  float prv[4];
#pragma unroll
  for (int j = 0; j < 4; ++j) {
    const float dd = (float)(l - (s0 + j));
    prv[j] = amp * __expf(dd * dd * cf);
  }
  float4 p4 = make_float4(prv[0], prv[1], prv[2], prv[3]);
  float4 s4 = make_float4(sval, sval, sval, sval);
  *(float4*)(outP + base + s0) = p4;   // global_store_b128
  *(float4*)(outT + base + s0) = s4;   // global_store_b128
}

extern "C" void kernel_launch(void* const* d_in, const int* in_sizes, int n_in,
                              void* d_out, int out_size, void* d_ws,
                              size_t ws_size, hipStream_t stream) {
  const float* Q     = (const float*)d_in[0];
  const float* K     = (const float*)d_in[1];
  const float* Vv    = (const float*)d_in[2];
  const float* sigma = (const float*)d_in[3];
  const float* sac   = (const float*)d_in[4];
  const int*   t     = (const int*)d_in[5];

  float* out = (float*)d_out;
  const size_t nV = (size_t)Bc * Lc * Hc * Dc;   //  2,097,152
  const size_t nS = (size_t)Bc * Hc * Lc * Lc;   // 33,554,432
  float* outV = out;
  float* outS = out + nV;
  float* outP = out + nV + nS;
  float* outT = out + nV + 2 * nS;

  attn_kernel<<<Bc * Hc * (Lc / 16), 256, 0, stream>>>(Q, K, Vv, outV, outS);
  prior_kernel<<<Bc * Hc * Lc, 256, 0, stream>>>(sigma, sac, t, outP, outT);
}